// GCN_user_filter_low_20727512170685
// MI455X (gfx1250) — compile-verified
//
#include <hip/hip_runtime.h>

#define U_N 100000
#define I_N 50000
#define DD  64
#define NNZ_N 3200000
#define B_N 16384
#define LAMBDA 0.001f

typedef __attribute__((ext_vector_type(2))) float v2f;
typedef __attribute__((ext_vector_type(8))) float v8f;

// Fire-and-forget f32 atomic add: guaranteed GLOBAL_ATOMIC_ADD_F32 (no return,
// STOREcnt-tracked, resolves at L2). s_endpgm's implicit wait-idle covers
// completion before the dispatch is considered done.
__device__ __forceinline__ void atomic_add_f32(float* p, float v) {
    asm volatile("global_atomic_add_f32 %0, %1, off"
                 :
                 : "v"(p), "v"(v)
                 : "memory");
}

// out[r*D+d] = x[r*D+d] * s[r]   (D = 64, so row = idx >> 6)
__global__ void k_scale_init(const float* __restrict__ x, const float* __restrict__ s,
                             float* __restrict__ out, int total) {
    int idx = blockIdx.x * blockDim.x + threadIdx.x;
    if (idx < total) out[idx] = x[idx] * s[idx >> 6];
}

// Edge-parallel SpMM scatter: one wave32 per edge, lane l handles float2 at col 2l.
// out[dst[e]] += vals[e] * x[src[e]]   (row vectors of length 64)
__global__ void k_spmm_scatter(const float* __restrict__ vals,
                               const int* __restrict__ src_idx,
                               const int* __restrict__ dst_idx,
                               const float* __restrict__ x,
                               float* __restrict__ out,
                               int nnz) {
    int lane  = threadIdx.x & 31;
    int wave  = (int)((blockIdx.x * blockDim.x + threadIdx.x) >> 5);
    int nwave = (int)((gridDim.x * blockDim.x) >> 5);
    for (int e = wave; e < nnz; e += nwave) {
        float v = vals[e];
        int   s = src_idx[e];
        int   d = dst_idx[e];
        v2f xv = *(const v2f*)(x + (size_t)s * DD + lane * 2);
        float* op = out + (size_t)d * DD + lane * 2;
        atomic_add_f32(op,     v * xv.x);
        atomic_add_f32(op + 1, v * xv.y);
    }
}

__global__ void k_relu(float* __restrict__ x, int n) {
    int idx = blockIdx.x * blockDim.x + threadIdx.x;
    if (idx < n) x[idx] = fmaxf(x[idx], 0.0f);
}

__global__ void k_zero(float* __restrict__ a) {
    if (threadIdx.x < 4) a[threadIdx.x] = 0.0f;
}

__device__ __forceinline__ float wave32_sum(float v) {
#pragma unroll
    for (int off = 16; off > 0; off >>= 1)
        v += __shfl_xor(v, off, 32);
    return v;
}

// One wave32 per 16 batch rows. Gathers gu = u0+g1u+g2u and gi = i0+g1i+g2i rows,
// computes pred via chained V_WMMA_F32_16X16X4_F32 (diagonal of user x item^T),
// and accumulates sum((pred - rating)^2) and sum(user^2)+sum(item^2).
__global__ void k_batch_pred(const float* __restrict__ u0, const float* __restrict__ g1u,
                             const float* __restrict__ g2u,
                             const float* __restrict__ i0, const float* __restrict__ g1i,
                             const float* __restrict__ g2i,
                             const int* __restrict__ user0, const int* __restrict__ item0,
                             const float* __restrict__ ratings,
                             float* __restrict__ accum) {
    int lane = threadIdx.x;         // blockDim.x == 32, EXEC all ones for WMMA
    int tile = blockIdx.x;          // B/16 tiles
    int m    = lane & 15;
    int half = lane >> 4;
    int b    = tile * 16 + m;
    int uu   = user0[b];
    int ii   = item0[b];

    const float* up0 = u0  + (size_t)uu * DD;
    const float* up1 = g1u + (size_t)uu * DD;
    const float* up2 = g2u + (size_t)uu * DD;
    const float* ip0 = i0  + (size_t)ii * DD;
    const float* ip1 = g1i + (size_t)ii * DD;
    const float* ip2 = g2i + (size_t)ii * DD;

    v8f c = {};
    float sq = 0.0f;
#pragma unroll
    for (int s = 0; s < 16; ++s) {
        int k = s * 4 + half * 2;       // lane's K-slot in A(16x4) / B(4x16) f32 layout
        v2f a, bb;
        a.x  = up0[k]     + up1[k]     + up2[k];
        a.y  = up0[k + 1] + up1[k + 1] + up2[k + 1];
        bb.x = ip0[k]     + ip1[k]     + ip2[k];
        bb.y = ip0[k + 1] + ip1[k + 1] + ip2[k + 1];
        sq += a.x * a.x + a.y * a.y + bb.x * bb.x + bb.y * bb.y;
        // D = A(16x4,f32) x B(4x16,f32) + C ; accumulate over K=64 in 16 steps
        c = __builtin_amdgcn_wmma_f32_16x16x4_f32(false, a, false, bb,
                                                  (short)0, c, false, false);
    }

    // Diagonal D[m][m] = pred[tile*16+m]:
    //   lanes 0..7   -> VGPR r = lane        (M = N = lane)
    //   lanes 24..31 -> VGPR r = lane - 24   (M = N = lane - 16)
    float diffsq = 0.0f;
    {
        int dm = (lane < 8) ? lane : ((lane >= 24) ? (lane - 16) : -1);
        int vr = (lane < 8) ? lane : ((lane >= 24) ? (lane - 24) : 0);
        union { v8f v; float f[8]; } cu; cu.v = c;
        float pred = cu.f[vr];
        if (dm >= 0) {
            float diff = pred - ratings[tile * 16 + dm];
            diffsq = diff * diff;
        }
    }

    // Wave-level reduction, then one atomic per wave per accumulator.
    float sq_sum   = wave32_sum(sq);      // covers full 16x64 user+item tile
    float diff_sum = wave32_sum(diffsq);  // 16 valid lanes, others contribute 0
    if (lane == 0) {
        atomic_add_f32(accum + 0, diff_sum);
        atomic_add_f32(accum + 1, sq_sum);
    }
}

__global__ void k_finalize(const float* __restrict__ accum, float* __restrict__ out) {
    if (threadIdx.x == 0) {
        float loss2 = accum[0] / (float)B_N;
        float l2    = LAMBDA * accum[1] / (float)(B_N * DD);
        out[0] = loss2;
        out[1] = loss2;
        out[2] = l2;
    }
}

extern "C" void kernel_launch(void* const* d_in, const int* in_sizes, int n_in,
                              void* d_out, int out_size, void* d_ws, size_t ws_size,
                              hipStream_t stream) {
    const float* u0      = (const float*)d_in[0];   // [U,D]
    const float* i0      = (const float*)d_in[1];   // [I,D]
    const float* ui_vals = (const float*)d_in[2];   // [NNZ]
    const float* d_i     = (const float*)d_in[3];   // [U,1]
    const float* d_j     = (const float*)d_in[4];   // [I,1]
    const float* ratings = (const float*)d_in[5];   // [B]
    const int*   ui_rows = (const int*)d_in[6];     // [NNZ]
    const int*   ui_cols = (const int*)d_in[7];     // [NNZ]
    const int*   user0   = (const int*)d_in[8];     // [B]
    const int*   item0   = (const int*)d_in[9];     // [B]
    float* out = (float*)d_out;

    float* ws  = (float*)d_ws;
    const size_t UD = (size_t)U_N * DD;   // 6,400,000
    const size_t ID = (size_t)I_N * DD;   // 3,200,000
    float* g1u   = ws;
    float* g2u   = g1u + UD;
    float* g1i   = g2u + UD;
    float* g2i   = g1i + ID;
    float* accum = g2i + ID;              // 4 floats

    const int TPB = 256;
    const int gU  = (int)((UD + TPB - 1) / TPB);
    const int gI  = (int)((ID + TPB - 1) / TPB);
    const int gE  = 8192;                 // 65536 waves over 3.2M edges
    const int gB  = B_N / 16;             // 1024 WMMA tiles

    k_zero<<<1, 32, 0, stream>>>(accum);

    // Layer 1: g1u = relu(spmm_ui(i0) + u0*d_i), g1i = relu(spmm_iu(u0) + i0*d_j)
    k_scale_init<<<gU, TPB, 0, stream>>>(u0, d_i, g1u, (int)UD);
    k_scale_init<<<gI, TPB, 0, stream>>>(i0, d_j, g1i, (int)ID);
    k_spmm_scatter<<<gE, TPB, 0, stream>>>(ui_vals, ui_cols, ui_rows, i0, g1u, NNZ_N);
    k_spmm_scatter<<<gE, TPB, 0, stream>>>(ui_vals, ui_rows, ui_cols, u0, g1i, NNZ_N);
    k_relu<<<gU, TPB, 0, stream>>>(g1u, (int)UD);
    k_relu<<<gI, TPB, 0, stream>>>(g1i, (int)ID);

    // Layer 2: g2u = relu(spmm_ui(g1i) + g1u*d_i), g2i = relu(spmm_iu(g1u) + g1i*d_j)
    k_scale_init<<<gU, TPB, 0, stream>>>(g1u, d_i, g2u, (int)UD);
    k_scale_init<<<gI, TPB, 0, stream>>>(g1i, d_j, g2i, (int)ID);
    k_spmm_scatter<<<gE, TPB, 0, stream>>>(ui_vals, ui_cols, ui_rows, g1i, g2u, NNZ_N);
    k_spmm_scatter<<<gE, TPB, 0, stream>>>(ui_vals, ui_rows, ui_cols, g1u, g2i, NNZ_N);
    k_relu<<<gU, TPB, 0, stream>>>(g2u, (int)UD);
    k_relu<<<gI, TPB, 0, stream>>>(g2i, (int)ID);

    // Batched dot products via V_WMMA_F32_16X16X4_F32 + scalar reductions
    k_batch_pred<<<gB, 32, 0, stream>>>(u0, g1u, g2u, i0, g1i, g2i,
                                        user0, item0, ratings, accum);
    k_finalize<<<1, 1, 0, stream>>>(accum, out);
}